// GraphEncoder_6287832121920
// MI455X (gfx1250) — compile-verified
//
#include <hip/hip_runtime.h>

#define N_NODES 20000
#define N_EDGES 320000
#define FD      64
#define NH      4
#define HF      256      // NH*FD
#define NGR     64
#define LATD    512
#define AGGD    832      // FD*(1+3*NH)
#define EL      (N_EDGES + N_NODES)   // edges + self loops

typedef float v2f __attribute__((ext_vector_type(2)));
typedef float v8f __attribute__((ext_vector_type(8)));

// ---- monotonic float<->uint key (for atomicMax on floats incl. negatives) ----
__device__ __forceinline__ unsigned f2key(float f) {
  unsigned b = __float_as_uint(f);
  return (b & 0x80000000u) ? ~b : (b | 0x80000000u);
}
__device__ __forceinline__ float key2f(unsigned k) {
  unsigned b = (k & 0x80000000u) ? (k ^ 0x80000000u) : ~k;
  return __uint_as_float(b);
}

// ---------------------------------------------------------------------------
// out[m,n] = (relu?)( sum_k X[m,k]*W[n,k] + bias[n] )  via V_WMMA_F32_16X16X4_F32
// One wave computes a 16x64 output tile: 4 accumulators share one A fragment
// per k-step. Software-pipelined: fragments for k+4 are loaded before the
// 4 WMMAs of step k issue, so loads overlap the WMMA chain instead of
// serializing on s_wait_loadcnt 0.
// A frag (16x4 f32): lanes 0-15 -> K={0,1}; lanes 16-31 -> K={2,3}; row = lane%16
// B frag (4x16 f32): same striping, col = lane%16 (B[k,n] = W[n,k], W row-major)
// C/D: VGPR v -> M=v (lanes 0-15) / M=v+8 (lanes 16-31), N = lane%16
// Requires: M % 16 == 0, K % 4 == 0 (and K >= 4), Ncols % 64 == 0, EXEC all ones.
// ---------------------------------------------------------------------------
__global__ __launch_bounds__(32) void gemm_xwt_wmma(
    const float* __restrict__ X, const float* __restrict__ W,
    const float* __restrict__ bias, float* __restrict__ out,
    int K, int Ncols, int applyRelu)
{
  const int colQuads = Ncols >> 6;            // 64 output columns per wave
  const int rowTile  = blockIdx.x / colQuads;
  const int colQuad  = blockIdx.x % colQuads;
  const int lane = threadIdx.x;
  const int half = lane >> 4;                 // 0: K pair {0,1}, 1: K pair {2,3}
  const int l16  = lane & 15;

  const float* xrow = X + (size_t)(rowTile * 16 + l16) * K;
  const float* w0 = W + (size_t)(colQuad * 64 +  0 + l16) * K;
  const float* w1 = W + (size_t)(colQuad * 64 + 16 + l16) * K;
  const float* w2 = W + (size_t)(colQuad * 64 + 32 + l16) * K;
  const float* w3 = W + (size_t)(colQuad * 64 + 48 + l16) * K;

  v8f c0 = {0.f,0.f,0.f,0.f,0.f,0.f,0.f,0.f};
  v8f c1 = c0, c2 = c0, c3 = c0;

  // prologue: fragments for k-step 0
  const int kk0 = half * 2;                   // 8B aligned (K multiple of 4)
  v2f a  = *(const v2f*)(xrow + kk0);
  v2f b0 = *(const v2f*)(w0 + kk0);
  v2f b1 = *(const v2f*)(w1 + kk0);
  v2f b2 = *(const v2f*)(w2 + kk0);
  v2f b3 = *(const v2f*)(w3 + kk0);

#pragma unroll 4
  for (int k0 = 4; k0 < K; k0 += 4) {
    const int kk = k0 + half * 2;
    // prefetch next k-step fragments before consuming current ones
    v2f an  = *(const v2f*)(xrow + kk);
    v2f bn0 = *(const v2f*)(w0 + kk);
    v2f bn1 = *(const v2f*)(w1 + kk);
    v2f bn2 = *(const v2f*)(w2 + kk);
    v2f bn3 = *(const v2f*)(w3 + kk);
    c0 = __builtin_amdgcn_wmma_f32_16x16x4_f32(false, a, false, b0, (short)0, c0, false, false);
    c1 = __builtin_amdgcn_wmma_f32_16x16x4_f32(false, a, false, b1, (short)0, c1, false, false);
    c2 = __builtin_amdgcn_wmma_f32_16x16x4_f32(false, a, false, b2, (short)0, c2, false, false);
    c3 = __builtin_amdgcn_wmma_f32_16x16x4_f32(false, a, false, b3, (short)0, c3, false, false);
    a = an; b0 = bn0; b1 = bn1; b2 = bn2; b3 = bn3;
  }
  // epilogue k-step
  c0 = __builtin_amdgcn_wmma_f32_16x16x4_f32(false, a, false, b0, (short)0, c0, false, false);
  c1 = __builtin_amdgcn_wmma_f32_16x16x4_f32(false, a, false, b1, (short)0, c1, false, false);
  c2 = __builtin_amdgcn_wmma_f32_16x16x4_f32(false, a, false, b2, (short)0, c2, false, false);
  c3 = __builtin_amdgcn_wmma_f32_16x16x4_f32(false, a, false, b3, (short)0, c3, false, false);

  const int mbase = rowTile * 16 + half * 8;
  const v8f* accs[4] = { &c0, &c1, &c2, &c3 };
#pragma unroll
  for (int sub = 0; sub < 4; ++sub) {
    const int ncol = colQuad * 64 + sub * 16 + l16;
    const float bv = bias ? bias[ncol] : 0.f;
    const v8f& c = *accs[sub];
#pragma unroll
    for (int v = 0; v < 8; ++v) {
      float val = c[v] + bv;
      if (applyRelu) val = fmaxf(val, 0.f);
      out[(size_t)(mbase + v) * Ncols + ncol] = val;
    }
  }
}

// ---------------------------------------------------------------------------
__global__ void zero_u32(unsigned* __restrict__ p, int n) {
  int i = blockIdx.x * 256 + threadIdx.x;
  if (i < n) p[i] = 0u;
}

// n0 = relu(street)*smask + relu(bfeat@bW^T + bb)*bmask   (street part precomputed)
__global__ void combine_nodes(const float* __restrict__ hs,
                              const float* __restrict__ bfeat,
                              const float* __restrict__ smask,
                              const float* __restrict__ bmask,
                              const float* __restrict__ bW,
                              const float* __restrict__ bb,
                              float* __restrict__ n0)
{
  int idx = blockIdx.x * 256 + threadIdx.x;
  if (idx >= N_NODES * FD) return;
  int n = idx >> 6, c = idx & 63;
  float acc = bb[c];
#pragma unroll
  for (int i = 0; i < 5; ++i) acc += bfeat[n * 5 + i] * bW[c * 5 + i];
  acc = fmaxf(acc, 0.f);
  n0[idx] = hs[idx] * smask[n] + acc * bmask[n];
}

// s[n,h] = <h[n,h,:], a_src[h,:]>,  d likewise
__global__ void attn_sd(const float* __restrict__ hm,
                        const float* __restrict__ asrc,
                        const float* __restrict__ adst,
                        float* __restrict__ s, float* __restrict__ d)
{
  int idx = blockIdx.x * 256 + threadIdx.x;
  if (idx >= N_NODES * NH) return;
  int n = idx >> 2, h = idx & 3;
  const float* hr = hm + (size_t)n * HF + h * FD;
  const float* as = asrc + h * FD;
  const float* ad = adst + h * FD;
  float ss = 0.f, dd = 0.f;
  for (int f = 0; f < FD; f += 4) {
    float4 hv = *(const float4*)(hr + f);
    float4 av = *(const float4*)(as + f);
    float4 dv = *(const float4*)(ad + f);
    ss += hv.x * av.x + hv.y * av.y + hv.z * av.z + hv.w * av.w;
    dd += hv.x * dv.x + hv.y * dv.y + hv.z * dv.z + hv.w * dv.w;
  }
  s[idx] = ss; d[idx] = dd;
}

__device__ __forceinline__ void edge_sd(const int* __restrict__ ei, int e,
                                        int& src, int& dst) {
  if (e < N_EDGES) { src = ei[e]; dst = ei[N_EDGES + e]; }
  else             { src = e - N_EDGES; dst = src; }        // self loop
}

__global__ void edge_max(const int* __restrict__ ei, const float* __restrict__ s,
                         const float* __restrict__ d, unsigned* __restrict__ mkey)
{
  int idx = blockIdx.x * 256 + threadIdx.x;
  if (idx >= EL * NH) return;
  int h = idx & 3, e = idx >> 2, src, dst;
  edge_sd(ei, e, src, dst);
  float x = s[src * NH + h] + d[dst * NH + h];
  x = x > 0.f ? x : 0.2f * x;                               // leaky_relu(0.2)
  atomicMax(&mkey[dst * NH + h], f2key(x));
}

__global__ void edge_exp(const int* __restrict__ ei, const float* __restrict__ s,
                         const float* __restrict__ d, const unsigned* __restrict__ mkey,
                         float* __restrict__ exb, float* __restrict__ den)
{
  int idx = blockIdx.x * 256 + threadIdx.x;
  if (idx >= EL * NH) return;
  int h = idx & 3, e = idx >> 2, src, dst;
  edge_sd(ei, e, src, dst);
  float x = s[src * NH + h] + d[dst * NH + h];
  x = x > 0.f ? x : 0.2f * x;
  float ex = expf(x - key2f(mkey[dst * NH + h]));
  exb[idx] = ex;
  atomicAdd(&den[dst * NH + h], ex);
}

__global__ void edge_alpha(const int* __restrict__ ei, const float* __restrict__ den,
                           float* __restrict__ exb)
{
  int idx = blockIdx.x * 256 + threadIdx.x;
  if (idx >= EL * NH) return;
  int h = idx & 3, e = idx >> 2, src, dst;
  edge_sd(ei, e, src, dst);
  exb[idx] = exb[idx] / den[dst * NH + h];
}

// acc[dst, c..c+3] += alpha[e, c/64] * h[src, c..c+3]   (one b128 load, 4 atomics)
__global__ void edge_scatter(const int* __restrict__ ei, const float* __restrict__ alpha,
                             const float* __restrict__ hm, float* __restrict__ acc)
{
  long idx = (long)blockIdx.x * 256 + threadIdx.x;
  if (idx >= (long)EL * (HF / 4)) return;
  int c4 = (int)(idx & 63);              // group of 4 channels
  int e  = (int)(idx >> 6);
  int src, dst; edge_sd(ei, e, src, dst);
  int c = c4 << 2;
  float w = alpha[e * NH + (c >> 6)];
  float4 hv = *(const float4*)(hm + (size_t)src * HF + c);
  float* ap = acc + (size_t)dst * HF + c;
  atomicAdd(ap + 0, w * hv.x);
  atomicAdd(ap + 1, w * hv.y);
  atomicAdd(ap + 2, w * hv.z);
  atomicAdd(ap + 3, w * hv.w);
}

__global__ void bias_relu(const float* __restrict__ acc, const float* __restrict__ b,
                          float* __restrict__ out)
{
  int idx = blockIdx.x * 256 + threadIdx.x;           // over N*HF/4
  if (idx >= N_NODES * (HF / 4)) return;
  int c = (idx & 63) << 2;
  float4 av = *(const float4*)(acc + (size_t)idx * 4);
  float4 bv = *(const float4*)(b + c);
  float4 r;
  r.x = fmaxf(av.x + bv.x, 0.f);
  r.y = fmaxf(av.y + bv.y, 0.f);
  r.z = fmaxf(av.z + bv.z, 0.f);
  r.w = fmaxf(av.w + bv.w, 0.f);
  *(float4*)(out + (size_t)idx * 4) = r;
}

// blockIdx.y = node; threads cover 832 concat channels
__global__ void pool_max(const float* __restrict__ n0, const float* __restrict__ n1,
                         const float* __restrict__ n2, const float* __restrict__ n3,
                         const int* __restrict__ batch, unsigned* __restrict__ gkey)
{
  int c = blockIdx.x * 256 + threadIdx.x;
  if (c >= AGGD) return;
  int n = blockIdx.y;
  float v;
  if (c < FD)              v = n0[(size_t)n * FD + c];
  else if (c < FD + HF)    v = n1[(size_t)n * HF + (c - FD)];
  else if (c < FD + 2*HF)  v = n2[(size_t)n * HF + (c - FD - HF)];
  else                     v = n3[(size_t)n * HF + (c - FD - 2 * HF)];
  atomicMax(&gkey[batch[n] * AGGD + c], f2key(v));
}

__global__ void decode_keys(const unsigned* __restrict__ k, float* __restrict__ f, int n) {
  int i = blockIdx.x * 256 + threadIdx.x;
  if (i < n) f[i] = key2f(k[i]);
}

// ---------------------------------------------------------------------------
extern "C" void kernel_launch(void* const* d_in, const int* in_sizes, int n_in,
                              void* d_out, int out_size, void* d_ws, size_t ws_size,
                              hipStream_t stream)
{
  const float* street_feature   = (const float*)d_in[0];
  const float* building_feature = (const float*)d_in[1];
  const float* street_mask      = (const float*)d_in[2];
  const float* building_mask    = (const float*)d_in[3];
  const int*   edge_index       = (const int*)  d_in[4];
  const int*   batch            = (const int*)  d_in[5];
  const float* street_W   = (const float*)d_in[6];
  const float* street_b   = (const float*)d_in[7];
  const float* building_W = (const float*)d_in[8];
  const float* building_b = (const float*)d_in[9];
  const float* W1 = (const float*)d_in[10];
  const float* a_src1 = (const float*)d_in[11];
  const float* a_dst1 = (const float*)d_in[12];
  const float* b1 = (const float*)d_in[13];
  const float* W2 = (const float*)d_in[14];
  const float* a_src2 = (const float*)d_in[15];
  const float* a_dst2 = (const float*)d_in[16];
  const float* b2 = (const float*)d_in[17];
  const float* W3 = (const float*)d_in[18];
  const float* a_src3 = (const float*)d_in[19];
  const float* a_dst3 = (const float*)d_in[20];
  const float* b3 = (const float*)d_in[21];
  const float* agg_W = (const float*)d_in[22];
  const float* agg_b = (const float*)d_in[23];
  const float* mu_W  = (const float*)d_in[24];
  const float* mu_b  = (const float*)d_in[25];
  const float* var_W = (const float*)d_in[26];
  const float* var_b = (const float*)d_in[27];

  // workspace carve-out (256B aligned)
  char* wsb = (char*)d_ws;
  size_t off = 0;
  auto take = [&](size_t bytes) -> char* {
    char* p = wsb + off;
    off += (bytes + 255) & ~(size_t)255;
    return p;
  };
  float*    n0   = (float*)take((size_t)N_NODES * FD * 4);
  float*    n1   = (float*)take((size_t)N_NODES * HF * 4);
  float*    n2   = (float*)take((size_t)N_NODES * HF * 4);
  float*    n3   = (float*)take((size_t)N_NODES * HF * 4);
  float*    hmat = (float*)take((size_t)N_NODES * HF * 4);
  float*    accb = (float*)take((size_t)N_NODES * HF * 4);
  float*    sbuf = (float*)take((size_t)N_NODES * NH * 4);
  float*    dbuf = (float*)take((size_t)N_NODES * NH * 4);
  unsigned* mkey = (unsigned*)take((size_t)N_NODES * NH * 4);
  float*    den  = (float*)take((size_t)N_NODES * NH * 4);
  float*    exb  = (float*)take((size_t)EL * NH * 4);
  unsigned* gkey = (unsigned*)take((size_t)NGR * AGGD * 4);
  float*    gbuf = (float*)take((size_t)NGR * AGGD * 4);
  float*    lat  = (float*)take((size_t)NGR * LATD * 4);

  auto cdiv = [](long a, long b) { return (int)((a + b - 1) / b); };

  // ---- node encode: relu(street @ street_W^T + street_b) via WMMA, then combine
  gemm_xwt_wmma<<<(N_NODES / 16) * (FD / 64), 32, 0, stream>>>(
      street_feature, street_W, street_b, hmat, 128, FD, 1);
  combine_nodes<<<cdiv((long)N_NODES * FD, 256), 256, 0, stream>>>(
      hmat, building_feature, street_mask, building_mask, building_W, building_b, n0);

  // ---- one GAT layer ----
  auto gat_layer = [&](const float* xin, int K, const float* Wm, const float* as,
                       const float* ad, const float* bv, float* nout) {
    gemm_xwt_wmma<<<(N_NODES / 16) * (HF / 64), 32, 0, stream>>>(
        xin, Wm, nullptr, hmat, K, HF, 0);
    attn_sd<<<cdiv((long)N_NODES * NH, 256), 256, 0, stream>>>(hmat, as, ad, sbuf, dbuf);
    zero_u32<<<cdiv((long)N_NODES * HF, 256), 256, 0, stream>>>((unsigned*)accb, N_NODES * HF);
    zero_u32<<<cdiv((long)N_NODES * NH, 256), 256, 0, stream>>>(mkey, N_NODES * NH);
    zero_u32<<<cdiv((long)N_NODES * NH, 256), 256, 0, stream>>>((unsigned*)den, N_NODES * NH);
    edge_max<<<cdiv((long)EL * NH, 256), 256, 0, stream>>>(edge_index, sbuf, dbuf, mkey);
    edge_exp<<<cdiv((long)EL * NH, 256), 256, 0, stream>>>(edge_index, sbuf, dbuf, mkey, exb, den);
    edge_alpha<<<cdiv((long)EL * NH, 256), 256, 0, stream>>>(edge_index, den, exb);
    edge_scatter<<<cdiv((long)EL * (HF / 4), 256), 256, 0, stream>>>(edge_index, exb, hmat, accb);
    bias_relu<<<cdiv((long)N_NODES * (HF / 4), 256), 256, 0, stream>>>(accb, bv, nout);
  };

  gat_layer(n0, FD, W1, a_src1, a_dst1, b1, n1);
  gat_layer(n1, HF, W2, a_src2, a_dst2, b2, n2);
  gat_layer(n2, HF, W3, a_src3, a_dst3, b3, n3);

  // ---- graph pooling (segment max over batch) ----
  zero_u32<<<cdiv((long)NGR * AGGD, 256), 256, 0, stream>>>(gkey, NGR * AGGD);
  {
    dim3 g(cdiv(AGGD, 256), N_NODES);
    pool_max<<<g, 256, 0, stream>>>(n0, n1, n2, n3, batch, gkey);
  }
  decode_keys<<<cdiv((long)NGR * AGGD, 256), 256, 0, stream>>>(gkey, gbuf, NGR * AGGD);

  // ---- VAE head: latent -> (mu, log_var), all WMMA ----
  float* outp = (float*)d_out;
  gemm_xwt_wmma<<<(NGR / 16) * (LATD / 64), 32, 0, stream>>>(gbuf, agg_W, agg_b, lat, AGGD, LATD, 0);
  gemm_xwt_wmma<<<(NGR / 16) * (LATD / 64), 32, 0, stream>>>(lat, mu_W,  mu_b,  outp,              LATD, LATD, 0);
  gemm_xwt_wmma<<<(NGR / 16) * (LATD / 64), 32, 0, stream>>>(lat, var_W, var_b, outp + NGR * LATD, LATD, LATD, 0);
}